// Router_80015240724581
// MI455X (gfx1250) — compile-verified
//
#include <hip/hip_runtime.h>
#include <hip/hip_bf16.h>
#include <cmath>

typedef __attribute__((ext_vector_type(2))) float v2f;
typedef __attribute__((ext_vector_type(4))) float v4f;
typedef __attribute__((ext_vector_type(8))) float v8f;

#define S_TOKENS 32768
#define DDIM     4096
#define NEXP     64
#define TOPK     8

#define KC        64                 // K-chunk staged in LDS
#define NCHUNK    (DDIM / KC)        // 64
#define KSTEPS    (KC / 4)           // 16 WMMA K-steps per chunk
#define ESTRIDE   65                 // padded expert stride (bank-conflict avoidance)
#define BUF_FLTS  (KSTEPS * ESTRIDE * 4)   // floats per LDS buffer

// -------- CDNA5 async global->LDS copy (ASYNCcnt-tracked), inline asm path ---
__device__ __forceinline__ void async_copy_b128(uint32_t lds_byte_off,
                                                const float* gsrc) {
  asm volatile("global_load_async_to_lds_b128 %0, %1, off"
               :: "v"(lds_byte_off), "v"(gsrc)
               : "memory");
}
__device__ __forceinline__ void wait_async_le4() {
  asm volatile("s_wait_asynccnt 0x4" ::: "memory");
}
__device__ __forceinline__ void wait_async_0() {
  asm volatile("s_wait_asynccnt 0x0" ::: "memory");
}

// Stage one K-chunk of W (64 experts x KC floats) into an LDS buffer.
// Copy i (0..1023): kstep = i & 15, e = i >> 4.  Consecutive lanes cover
// consecutive ksteps of one expert row -> coalesced 256B global segments.
// LDS layout: [kstep][e (stride 65)][4 floats of K].
__device__ __forceinline__ void stage_chunk(const float* __restrict__ W,
                                            int k0, float* bsbuf, int tid) {
#pragma unroll
  for (int j = 0; j < 4; ++j) {
    const int i     = tid + 256 * j;        // 0..1023
    const int kstep = i & (KSTEPS - 1);
    const int e     = i >> 4;
    const float*   gsrc = W + (size_t)e * DDIM + k0 + kstep * 4;
    const uint32_t ldso =
        (uint32_t)(size_t)(bsbuf + (kstep * ESTRIDE + e) * 4);
    async_copy_b128(ldso, gsrc);
  }
}

// ---------------------------------------------------------------------------
// Kernel 1: router logits GEMM.  One wave32 computes a 16-token x 64-expert
// tile with V_WMMA_F32_16X16X4_F32 (f32 preserved end-to-end: logits feed a
// top-k, so no downcast).  x (512 MB) is the only HBM stream and is read
// exactly once; W (1 MB) is staged per-block into double-buffered LDS via
// async copies overlapped with compute.
// ---------------------------------------------------------------------------
__global__ __launch_bounds__(256) void router_gemm_kernel(
    const float* __restrict__ x, const float* __restrict__ W,
    const float* __restrict__ b, float* __restrict__ logits)
{
  __shared__ float Bs[2][BUF_FLTS];

  const int tid  = threadIdx.x;
  const int lane = tid & 31;
  const int wave = tid >> 5;
  const int half = lane >> 4;          // 0 or 1
  const int lm   = lane & 15;
  const int row0 = (blockIdx.x * 8 + wave) * 16;

  // A fragment (16x4 f32): lane L<16 -> x[row0+L, k..k+1],
  //                        lane L>=16 -> x[row0+(L-16), k+2..k+3]
  const float* aptr = x + (size_t)(row0 + lm) * DDIM + 2 * half;

  v8f acc0 = {}, acc1 = {}, acc2 = {}, acc3 = {};

  // prologue: stage chunk 0
  stage_chunk(W, 0, &Bs[0][0], tid);

  for (int c = 0; c < NCHUNK; ++c) {
    const int buf = c & 1;
    if (c + 1 < NCHUNK) {
      stage_chunk(W, (c + 1) * KC, &Bs[buf ^ 1][0], tid);
      wait_async_le4();               // chunk c's 4 copies retired (in-order)
    } else {
      wait_async_0();
    }
    __syncthreads();                  // all waves' copies visible

    const int k0 = c * KC;
    // B fragment base for this lane: [kstep][16*eg + lm][2*half .. +1]
    const float* bb = &Bs[buf][lm * 4 + 2 * half];

    // pull next x cachelines toward the WGP while we compute
    __builtin_prefetch(aptr + k0 + 2 * KC, 0, 1);

#pragma unroll 8
    for (int t = 0; t < KSTEPS; ++t) {
      v2f a  = *(const v2f*)(aptr + k0 + t * 4);
      const float* bt = bb + t * (ESTRIDE * 4);
      v2f b0 = *(const v2f*)(bt);            // experts  0..15  (256B apart ->
      v2f b1 = *(const v2f*)(bt + 64);       // experts 16..31   ds_load_2addr)
      v2f b2 = *(const v2f*)(bt + 128);      // experts 32..47
      v2f b3 = *(const v2f*)(bt + 192);      // experts 48..63
      acc0 = __builtin_amdgcn_wmma_f32_16x16x4_f32(false, a, false, b0, (short)0, acc0, false, false);
      acc1 = __builtin_amdgcn_wmma_f32_16x16x4_f32(false, a, false, b1, (short)0, acc1, false, false);
      acc2 = __builtin_amdgcn_wmma_f32_16x16x4_f32(false, a, false, b2, (short)0, acc2, false, false);
      acc3 = __builtin_amdgcn_wmma_f32_16x16x4_f32(false, a, false, b3, (short)0, acc3, false, false);
    }
    __syncthreads();                  // buf free for chunk c+2 staging
  }

  // C/D layout (ISA 7.12.2): VGPR j, lanes 0-15 -> M=j, N=lane;
  //                          lanes 16-31 -> M=j+8, N=lane-16.
  const float bias0 = b[ 0 + lm];
  const float bias1 = b[16 + lm];
  const float bias2 = b[32 + lm];
  const float bias3 = b[48 + lm];
#pragma unroll
  for (int j = 0; j < 8; ++j) {
    const int token = row0 + j + 8 * half;
    float* o = logits + (size_t)token * NEXP + lm;
    o[ 0] = acc0[j] + bias0;
    o[16] = acc1[j] + bias1;
    o[32] = acc2[j] + bias2;
    o[48] = acc3[j] + bias3;
  }
}

// ---------------------------------------------------------------------------
// Kernel 2: per-token top-8 selection, softmax over selected, scatter mask.
// One thread per token; 64 logits staged in LDS for the dynamic-index
// knockout argmax (strict '>' => lowest-index tie-break, matches lax.top_k).
// ---------------------------------------------------------------------------
__global__ __launch_bounds__(128) void router_topk_kernel(
    const float* __restrict__ logits, float* __restrict__ out_idx,
    float* __restrict__ out_w, float* __restrict__ out_mask)
{
  __shared__ float lds[128 * NEXP];
  const int t = blockIdx.x * 128 + threadIdx.x;
  float* my = lds + threadIdx.x * NEXP;
  const float* src = logits + (size_t)t * NEXP;

#pragma unroll
  for (int e = 0; e < NEXP; e += 4) {
    v4f v = *(const v4f*)(src + e);
    my[e + 0] = v.x;
    my[e + 1] = v.y;
    my[e + 2] = v.z;
    my[e + 3] = v.w;
  }

  int   idx[TOPK];
  float val[TOPK];
#pragma unroll
  for (int j = 0; j < TOPK; ++j) {
    float best = -INFINITY;
    int   bi   = 0;
    for (int e = 0; e < NEXP; ++e) {
      float v = my[e];
      if (v > best) { best = v; bi = e; }
    }
    idx[j] = bi;
    val[j] = best;
    my[bi] = -INFINITY;   // knock out for next pass
  }

  const float m = val[0];
  float s = 0.0f;
  float w[TOPK];
#pragma unroll
  for (int j = 0; j < TOPK; ++j) { w[j] = __expf(val[j] - m); s += w[j]; }
  const float inv = 1.0f / s;

#pragma unroll
  for (int j = 0; j < TOPK; ++j) {
    out_idx[(size_t)t * TOPK + j] = (float)idx[j];
    out_w  [(size_t)t * TOPK + j] = w[j] * inv;
  }

  for (int e = 0; e < NEXP; ++e) {
    float mv = 0.0f;
#pragma unroll
    for (int j = 0; j < TOPK; ++j) mv = (idx[j] == e) ? 1.0f : mv;
    out_mask[(size_t)t * NEXP + e] = mv;
  }
}

extern "C" void kernel_launch(void* const* d_in, const int* in_sizes, int n_in,
                              void* d_out, int out_size, void* d_ws, size_t ws_size,
                              hipStream_t stream) {
  const float* x = (const float*)d_in[0];
  const float* W = (const float*)d_in[1];
  const float* b = (const float*)d_in[2];

  float* out    = (float*)d_out;
  float* logits = out;                                    // [S, 64]
  float* oidx   = logits + (size_t)S_TOKENS * NEXP;       // [S, 8] (as float)
  float* ow     = oidx   + (size_t)S_TOKENS * TOPK;       // [S, 8]
  float* omask  = ow     + (size_t)S_TOKENS * TOPK;       // [S, 64]

  // 256 threads = 8 waves; each wave -> 16 tokens => 128 tokens/block
  router_gemm_kernel<<<dim3(S_TOKENS / 128), dim3(256), 0, stream>>>(x, W, b, logits);
  // 1 thread per token, 128 tokens/block
  router_topk_kernel<<<dim3(S_TOKENS / 128), dim3(128), 0, stream>>>(logits, oidx, ow, omask);
}